// STConvBlock_80075370266890
// MI455X (gfx1250) — compile-verified
//
#include <hip/hip_runtime.h>
#include <hip/hip_bf16.h>
#include <math.h>

// Problem constants (from reference): B=64, N=17, T=512, C=64, H=4, DK=16
#define BB   64
#define NNODE 17
#define TT   512
#define CCH  64
#define NHEAD 4
#define BT_  (BB * TT)          // 32768
#define ROWS (BT_ * NNODE)      // 557056  (= 16 * 34816, no tail)

typedef __attribute__((ext_vector_type(2))) float v2f;
typedef __attribute__((ext_vector_type(8))) float v8f;

// -------------------------------------------------------------------------
// Kernel 1: H[r, o] = sum_c h0[r, c] * W[o, c]   (r = bt*17 + n)
// fp32 WMMA 16x16x4. One wave -> 16 rows x 64 cols, K=64 -> 64 wmma ops.
// h0[bt, n, c] = x[b, n, t, c] with bt = b*512 + t (x layout IS h0 row layout
// after index remap, so we address x directly per row).
// -------------------------------------------------------------------------
__global__ __launch_bounds__(256) void k1_gemm(const float* __restrict__ x,
                                               const float* __restrict__ W,
                                               float* __restrict__ Hout) {
    __shared__ float Wl[64 * 64];
    for (int i = threadIdx.x; i < 64 * 64; i += 256) Wl[i] = W[i];
    __syncthreads();

    const int wv   = threadIdx.x >> 5;
    const int lane = threadIdx.x & 31;
    const int half = lane >> 4;       // 0: K+0/K+1, 1: K+2/K+3
    const int lm   = lane & 15;
    const int m0   = (blockIdx.x * 8 + wv) * 16;

    // A-fragment row for this lane
    const int r  = m0 + lm;
    const int bt = r / 17;
    const int n  = r - bt * 17;
    const int b  = bt >> 9;           // /512
    const int t  = bt & 511;
    const float* px = x + ((((size_t)b * NNODE + n) * TT + t) * CCH);
    const int koff = half * 2;

    v8f acc0 = {}; v8f acc1 = {}; v8f acc2 = {}; v8f acc3 = {};

#pragma unroll
    for (int kk = 0; kk < 64; kk += 4) {
        v2f av; av.x = px[kk + koff]; av.y = px[kk + koff + 1];
        v2f b0, b1, b2, b3;
        b0.x = Wl[(0 * 16 + lm) * 64 + kk + koff]; b0.y = Wl[(0 * 16 + lm) * 64 + kk + koff + 1];
        b1.x = Wl[(1 * 16 + lm) * 64 + kk + koff]; b1.y = Wl[(1 * 16 + lm) * 64 + kk + koff + 1];
        b2.x = Wl[(2 * 16 + lm) * 64 + kk + koff]; b2.y = Wl[(2 * 16 + lm) * 64 + kk + koff + 1];
        b3.x = Wl[(3 * 16 + lm) * 64 + kk + koff]; b3.y = Wl[(3 * 16 + lm) * 64 + kk + koff + 1];
        acc0 = __builtin_amdgcn_wmma_f32_16x16x4_f32(false, av, false, b0, (short)0, acc0, false, false);
        acc1 = __builtin_amdgcn_wmma_f32_16x16x4_f32(false, av, false, b1, (short)0, acc1, false, false);
        acc2 = __builtin_amdgcn_wmma_f32_16x16x4_f32(false, av, false, b2, (short)0, acc2, false, false);
        acc3 = __builtin_amdgcn_wmma_f32_16x16x4_f32(false, av, false, b3, (short)0, acc3, false, false);
    }

    // C/D layout: VGPR i -> row m0+i (lanes 0-15) / m0+i+8 (lanes 16-31), col = lane%16
#pragma unroll
    for (int i = 0; i < 8; ++i) {
        float* op = Hout + (size_t)(m0 + i + half * 8) * 64 + lm;
        op[0]  = acc0[i];
        op[16] = acc1[i];
        op[32] = acc2[i];
        op[48] = acc3[i];
    }
}

// -------------------------------------------------------------------------
// Kernel 2: per (b,t): s/t projections, e=lrelu(s_i+t_j), +adj bias,
// softmax over j, scale_j = sum_i alpha_ij, G = H*scale + h0.
// One wave per bt (17 active lanes). G written in x layout [B,N,T,C].
// -------------------------------------------------------------------------
__global__ __launch_bounds__(256) void k2_attn(const float* __restrict__ Hbuf,
                                               const float* __restrict__ x,
                                               const int*   __restrict__ adj,
                                               const float* __restrict__ avec,
                                               float* __restrict__ G) {
    __shared__ float sv[8][NNODE][NHEAD];
    __shared__ float tv[8][NNODE][NHEAD];
    __shared__ float al[8][NNODE][NNODE];

    const int wv   = threadIdx.x >> 5;
    const int lane = threadIdx.x & 31;
    const int bt   = blockIdx.x * 8 + wv;
    const int b    = bt >> 9;
    const int t    = bt & 511;

    if (lane < NNODE) {
        const float* hr = Hbuf + ((size_t)bt * NNODE + lane) * CCH;
#pragma unroll
        for (int h = 0; h < NHEAD; ++h) {
            float s = 0.f, tt = 0.f;
#pragma unroll
            for (int d = 0; d < 16; ++d) {
                const float hv = hr[h * 16 + d];
                s  += hv * avec[d];
                tt += hv * avec[16 + d];
            }
            sv[wv][lane][h] = s;
            tv[wv][lane][h] = tt;
        }
    }
    __syncthreads();

    for (int h = 0; h < NHEAD; ++h) {
        if (lane < NNODE) {
            const int i = lane;
            const float si = sv[wv][i][h];
            float row[NNODE];
            float mx = -3.0e38f;
#pragma unroll
            for (int j = 0; j < NNODE; ++j) {
                float e = si + tv[wv][j][h];
                e = (e > 0.f) ? e : 0.2f * e;                 // leaky_relu(0.2)
                if (adj[(size_t)b * (NNODE * NNODE) + i * NNODE + j] == 0) e += -1e9f;
                row[j] = e;
                mx = fmaxf(mx, e);
            }
            float sum = 0.f;
#pragma unroll
            for (int j = 0; j < NNODE; ++j) { row[j] = __expf(row[j] - mx); sum += row[j]; }
            const float inv = 1.f / sum;
#pragma unroll
            for (int j = 0; j < NNODE; ++j) al[wv][i][j] = row[j] * inv;
        }
        __syncthreads();
        if (lane < NNODE) {
            const int j = lane;
            float sc = 0.f;
#pragma unroll
            for (int i = 0; i < NNODE; ++i) sc += al[wv][i][j];
            const float* hj = Hbuf + ((size_t)bt * NNODE + j) * CCH;
            const size_t base = (((size_t)b * NNODE + j) * TT + t) * CCH;
#pragma unroll
            for (int d = 0; d < 16; ++d) {
                const int c = h * 16 + d;
                G[base + c] = hj[c] * sc + x[base + c];
            }
        }
        __syncthreads();
    }
}

// -------------------------------------------------------------------------
// Kernel 3: per (b,n): flat [T*C] block reinterpreted as 64 channels x 512.
// Two causal depthwise convs (k=3, dil 1 then 2) + eval-BN + exact GELU.
// out[b,n,t,c] = y2[ch=c, time=t] + x[b, m%17, t, m/17] (m = n*64+c).
// One wave per (bn, channel); lane computes 16 timesteps in registers.
// -------------------------------------------------------------------------
__global__ __launch_bounds__(256) void k3_tcn(const float* __restrict__ G,
                                              const float* __restrict__ x,
                                              const float* __restrict__ w1,
                                              const float* __restrict__ g1,
                                              const float* __restrict__ be1v,
                                              const float* __restrict__ w2,
                                              const float* __restrict__ g2,
                                              const float* __restrict__ be2v,
                                              float* __restrict__ out) {
    const int wv   = threadIdx.x >> 5;
    const int lane = threadIdx.x & 31;
    const int ch   = blockIdx.x * 8 + wv;     // [0, 1088*64)
    const int bn   = ch >> 6;
    const int c    = ch & 63;
    const int b    = bn / 17;
    const int n    = bn - b * 17;

    const float* gp = G + (size_t)bn * (TT * CCH) + (size_t)c * TT;
    const int tb = lane * 16;

    const float invbn = 0.999995000037f;       // 1/sqrt(1+1e-5)
    const float ISQ2  = 0.70710678118654752f;

    float vv[22];
#pragma unroll
    for (int k = 0; k < 22; ++k) {
        const int idx = tb - 6 + k;
        vv[k] = (idx >= 0) ? gp[idx] : 0.f;    // causal zero pad (conv1 input)
    }

    const float w10 = w1[c * 3 + 0], w11 = w1[c * 3 + 1], w12 = w1[c * 3 + 2];
    const float ga1 = g1[c], bb1 = be1v[c];

    float uu[20];                              // times tb-4 .. tb+15
#pragma unroll
    for (int i = 0; i < 20; ++i) {
        const int tt = tb - 4 + i;
        float y = w10 * vv[i] + w11 * vv[i + 1] + w12 * vv[i + 2];
        y = y * invbn * ga1 + bb1;
        y = 0.5f * y * (1.f + erff(y * ISQ2));
        uu[i] = (tt >= 0) ? y : 0.f;           // causal zero pad (conv2 input)
    }

    const float w20 = w2[c * 3 + 0], w21 = w2[c * 3 + 1], w22 = w2[c * 3 + 2];
    const float ga2 = g2[c], bb2 = be2v[c];

    // residual scramble: m = n*64 + c -> x[b, m%17, t, m/17]
    const int m  = n * 64 + c;
    const int c2 = m / 17;
    const int n2 = m - c2 * 17;
    const float* rp = x   + ((((size_t)b * NNODE + n2) * TT + tb) * CCH + c2);
    float*       op = out + ((((size_t)b * NNODE + n ) * TT + tb) * CCH + c );

#pragma unroll
    for (int q = 0; q < 16; ++q) {
        float y = w20 * uu[q] + w21 * uu[q + 2] + w22 * uu[q + 4];
        y = y * invbn * ga2 + bb2;
        y = 0.5f * y * (1.f + erff(y * ISQ2));
        op[(size_t)q * CCH] = y + rp[(size_t)q * CCH];
    }
}

// -------------------------------------------------------------------------
// Launch. Workspace: H (557056*64 f32 = 142.6 MB) + G (same) = 285 MB.
// -------------------------------------------------------------------------
extern "C" void kernel_launch(void* const* d_in, const int* in_sizes, int n_in,
                              void* d_out, int out_size, void* d_ws, size_t ws_size,
                              hipStream_t stream) {
    const float* x    = (const float*)d_in[0];
    const int*   adj  = (const int*)  d_in[1];
    const float* W    = (const float*)d_in[2];
    const float* avec = (const float*)d_in[3];
    const float* w1   = (const float*)d_in[4];
    const float* g1   = (const float*)d_in[5];
    const float* b1   = (const float*)d_in[6];
    const float* w2   = (const float*)d_in[7];
    const float* g2   = (const float*)d_in[8];
    const float* b2   = (const float*)d_in[9];
    float* out = (float*)d_out;

    float* Hbuf = (float*)d_ws;
    float* Gbuf = Hbuf + (size_t)ROWS * CCH;

    // 557056 rows / 16 per wave / 8 waves per block = 4352 blocks
    k1_gemm<<<ROWS / 16 / 8, 256, 0, stream>>>(x, W, Hbuf);
    // 32768 bt / 8 waves per block
    k2_attn<<<BT_ / 8, 256, 0, stream>>>(Hbuf, x, adj, avec, Gbuf);
    // 1088*64 channels / 8 waves per block
    k3_tcn<<<(BB * NNODE * CCH) / 8, 256, 0, stream>>>(Gbuf, x, w1, g1, b1, w2, g2, b2, out);
}